// GATLayer_26328149525023
// MI455X (gfx1250) — compile-verified
//
#include <hip/hip_runtime.h>
#include <hip/hip_bf16.h>

#define NEG_SLOPE 0.01f
#define USE_ASYNC_LDS 1

typedef float v2f __attribute__((ext_vector_type(2)));
typedef float v8f __attribute__((ext_vector_type(8)));

#ifndef __has_builtin
#define __has_builtin(x) 0
#endif
#if __has_builtin(__builtin_amdgcn_wmma_f32_16x16x4_f32)
#define HAVE_WMMA_F32X4 1
#else
#define HAVE_WMMA_F32X4 0
#endif

// ---------------------------------------------------------------------------
// Kernel 1: z = h @ W_fc   (M=50000, K=256, N=64)
// One block = 16-row M panel staged in LDS (async global->LDS b128, ASYNCcnt);
// 4 waves each own a 16-col N tile; 64 x V_WMMA_F32_16X16X4_F32 per wave.
// ---------------------------------------------------------------------------
__global__ __launch_bounds__(128) void gat_gemm_z(const float* __restrict__ h,
                                                  const float* __restrict__ W,
                                                  float* __restrict__ z) {
  __shared__ float As[16 * 256];  // 16 KB A panel
  const int tid = threadIdx.x;
  const int m_base = blockIdx.x * 16;

  // Stage 16 rows x 256 cols = 1024 float4 into LDS.
  const float4* h4 = (const float4*)(h + (size_t)m_base * 256);
  float4* As4 = (float4*)As;
#if USE_ASYNC_LDS && defined(__gfx1250__)
#pragma unroll
  for (int i = 0; i < 8; ++i) {
    const float4* g = h4 + tid + i * 128;
    // Low 32 bits of a flat LDS pointer == wave-relative LDS byte address.
    unsigned l = (unsigned)(uintptr_t)(As4 + tid + i * 128);
    asm volatile("global_load_async_to_lds_b128 %0, %1, off"
                 :
                 : "v"(l), "v"(g)
                 : "memory");
  }
  asm volatile("s_wait_asynccnt 0x0" ::: "memory");
#else
#pragma unroll
  for (int i = 0; i < 8; ++i) As4[tid + i * 128] = h4[tid + i * 128];
#endif
  __syncthreads();

  const int wave = tid >> 5;
  const int lane = tid & 31;
  const int n_base = wave * 16;
  const int half = lane & 15;          // M row (A) or N col (B/C)
  const int khalf = (lane >> 4) << 1;  // 0 for lanes 0-15, 2 for lanes 16-31

  v8f c = {0.f, 0.f, 0.f, 0.f, 0.f, 0.f, 0.f, 0.f};

#if HAVE_WMMA_F32X4
  const float* Arow = &As[half * 256];
#pragma unroll 4
  for (int k0 = 0; k0 < 256; k0 += 4) {
    v2f a, b;
    // 32-bit A 16x4: VGPR0 = K=k0 (lanes 0-15) / k0+2 (lanes 16-31); VGPR1 = +1
    a.x = Arow[k0 + khalf];
    a.y = Arow[k0 + khalf + 1];
    // 32-bit B 4x16: VGPR0 = row K=k0 (lanes 0-15) / k0+2 (lanes 16-31); VGPR1 = +1
    b.x = W[(k0 + khalf) * 64 + n_base + half];
    b.y = W[(k0 + khalf + 1) * 64 + n_base + half];
    c = __builtin_amdgcn_wmma_f32_16x16x4_f32(false, a, false, b, (short)0, c,
                                              false, false);
  }
#else
  // Scalar fallback (also what the host pass parses): same tile ownership.
  for (int r = 0; r < 8; ++r) {
    const int m = r + ((lane >> 4) << 3);
    const int n = n_base + half;
    float acc = 0.f;
    for (int k = 0; k < 256; ++k) acc += As[m * 256 + k] * W[k * 64 + n];
    c[r] = acc;
  }
#endif

  // C/D layout: VGPR r -> M = r + 8*(lane>=16), N = lane&15 (+ n_base)
#pragma unroll
  for (int r = 0; r < 8; ++r) {
    const int m = r + ((lane >> 4) << 3);
    z[(size_t)(m_base + m) * 64 + n_base + half] = c[r];
  }
}

// ---------------------------------------------------------------------------
// Kernel 2: s_l = z @ a_attn[:64], s_r = z @ a_attn[64:]  (one wave per node)
// ---------------------------------------------------------------------------
__global__ __launch_bounds__(256) void gat_scores(const float* __restrict__ z,
                                                  const float* __restrict__ a_attn,
                                                  float* __restrict__ s_l,
                                                  float* __restrict__ s_r,
                                                  int n_nodes) {
  const int node = (int)((blockIdx.x * blockDim.x + threadIdx.x) >> 5);
  const int lane = threadIdx.x & 31;
  if (node >= n_nodes) return;
  const float* zr = z + (size_t)node * 64;
  const float z0 = zr[lane];
  const float z1 = zr[lane + 32];
  float pl = z0 * a_attn[lane] + z1 * a_attn[lane + 32];
  float pr = z0 * a_attn[64 + lane] + z1 * a_attn[96 + lane];
#pragma unroll
  for (int off = 16; off > 0; off >>= 1) {
    pl += __shfl_xor(pl, off, 32);
    pr += __shfl_xor(pr, off, 32);
  }
  if (lane == 0) {
    s_l[node] = pl;
    s_r[node] = pr;
  }
}

// ---------------------------------------------------------------------------
// Kernel 3: init m = -inf, den = 0, h_out = 0
// ---------------------------------------------------------------------------
__global__ void gat_init(float* __restrict__ m, float* __restrict__ den,
                         float* __restrict__ h_out, int n_nodes) {
  const int i = blockIdx.x * blockDim.x + threadIdx.x;
  if (i < n_nodes) {
    m[i] = -__builtin_inff();
    den[i] = 0.f;
  }
  if (i < n_nodes * 64) h_out[i] = 0.f;
}

// ---------------------------------------------------------------------------
// Kernel 4: e = leaky_relu(s_l[src] + s_r[dst]); segment_max via ordered-int
// atomics (monotone bit mapping, no CAS loop).
// ---------------------------------------------------------------------------
__global__ void gat_edge_max(const int* __restrict__ src,
                             const int* __restrict__ dst,
                             const float* __restrict__ s_l,
                             const float* __restrict__ s_r,
                             float* __restrict__ ebuf, float* __restrict__ m,
                             int n_edges) {
  const int i = blockIdx.x * blockDim.x + threadIdx.x;
  if (i >= n_edges) return;
  const int d = dst[i];
  float e = s_l[src[i]] + s_r[d];
  e = e > 0.f ? e : NEG_SLOPE * e;
  ebuf[i] = e;
  if (e >= 0.f)
    atomicMax((int*)&m[d], __float_as_int(e));
  else
    atomicMin((unsigned int*)&m[d], (unsigned int)__float_as_int(e));
}

// ---------------------------------------------------------------------------
// Kernel 5: ex = exp(e - m[dst]); den[dst] += ex  (global_atomic_add_f32)
// ---------------------------------------------------------------------------
__global__ void gat_edge_exp(const int* __restrict__ dst,
                             float* __restrict__ ebuf,
                             const float* __restrict__ m,
                             float* __restrict__ den, int n_edges) {
  const int i = blockIdx.x * blockDim.x + threadIdx.x;
  if (i >= n_edges) return;
  const int d = dst[i];
  const float ex = __expf(ebuf[i] - m[d]);
  ebuf[i] = ex;
  atomicAdd(&den[d], ex);
}

// ---------------------------------------------------------------------------
// Kernel 6: h_out[dst] += (ex/den[dst]) * z[src]  (one wave per edge, float2
// per lane -> coalesced 256B gather + 64 f32 atomics per edge)
// ---------------------------------------------------------------------------
__global__ __launch_bounds__(256) void gat_scatter(
    const int* __restrict__ src, const int* __restrict__ dst,
    const float* __restrict__ ebuf, const float* __restrict__ den,
    const float* __restrict__ z, float* __restrict__ h_out, int n_edges) {
  const int e = (int)((blockIdx.x * blockDim.x + threadIdx.x) >> 5);
  const int lane = threadIdx.x & 31;
  if (e >= n_edges) return;
  const int s = src[e];
  const int d = dst[e];
  const float alpha = ebuf[e] / den[d];
  const float2 v = ((const float2*)(z + (size_t)s * 64))[lane];
  float* out = h_out + (size_t)d * 64 + lane * 2;
  atomicAdd(out, alpha * v.x);
  atomicAdd(out + 1, alpha * v.y);
}

// ---------------------------------------------------------------------------
extern "C" void kernel_launch(void* const* d_in, const int* in_sizes, int n_in,
                              void* d_out, int out_size, void* d_ws,
                              size_t ws_size, hipStream_t stream) {
  const float* h = (const float*)d_in[0];
  const int* esrc = (const int*)d_in[1];
  const int* edst = (const int*)d_in[2];
  const float* W = (const float*)d_in[3];
  const float* a_attn = (const float*)d_in[4];
  float* h_out = (float*)d_out;

  const int in_dim = 256;
  const int n_nodes = in_sizes[0] / in_dim;  // 50000
  const int n_edges = in_sizes[1];           // 800000

  // Workspace layout (floats): z | s_l | s_r | m | den | ebuf  (~16.8 MB)
  float* ws = (float*)d_ws;
  float* z = ws;
  float* s_l = z + (size_t)n_nodes * 64;
  float* s_r = s_l + n_nodes;
  float* m = s_r + n_nodes;
  float* den = m + n_nodes;
  float* ebuf = den + n_nodes;

  const int mblocks = (n_nodes + 15) / 16;  // 3125 (exact)
  gat_gemm_z<<<mblocks, 128, 0, stream>>>(h, W, z);
  gat_scores<<<(n_nodes + 7) / 8, 256, 0, stream>>>(z, a_attn, s_l, s_r,
                                                    n_nodes);
  gat_init<<<(n_nodes * 64 + 255) / 256, 256, 0, stream>>>(m, den, h_out,
                                                           n_nodes);
  gat_edge_max<<<(n_edges + 255) / 256, 256, 0, stream>>>(esrc, edst, s_l, s_r,
                                                          ebuf, m, n_edges);
  gat_edge_exp<<<(n_edges + 255) / 256, 256, 0, stream>>>(edst, ebuf, m, den,
                                                          n_edges);
  gat_scatter<<<(size_t)n_edges * 32 / 256 + 1, 256, 0, stream>>>(
      esrc, edst, ebuf, den, z, h_out, n_edges);
}